// InteractLayerVec_62740882260437
// MI455X (gfx1250) — compile-verified
//
#include <hip/hip_runtime.h>
#include <hip/hip_bf16.h>
#include <math.h>

// ---------------------------------------------------------------------------
// HIP-NN interaction layer for MI455X (gfx1250, wave32, WMMA).
//
// Phase A (envsum_wmma_kernel): one workgroup (5 waves) per atom.
//   The segmented reduction IS a GEMM:  env_all[80,80] = S_all @ F
//   where S_all[m, p] stacks {sense, sense*u0, sense*u1, sense*u2} (m=0..79)
//   and F[p, f] are gathered neighbor features. Pairs are staged in LDS in
//   chunks of 64 (transposed, zero-padded), and each wave owns one 16-row
//   M-tile, accumulating 5 N-tiles in registers via V_WMMA_F32_16X16X4_F32.
//   Output rows 0-19 -> env, 20-39 -> envv0, 40-59 -> envv1, 60-79 -> envv2.
//
// Phase B (gemm_kernel): one wave per 16x16 output tile.
//   D = env @ W  with W[k=s*80+f, o] = int_weights[s, o, f], K=1600 WMMA loop
//   (4 accumulators: scalar + 3 vector components) plus K=80 self-interaction,
//   epilogue applies sqrt(sum_c fvec_c^2 + 1e-30)*vecscales + bias.
// ---------------------------------------------------------------------------

#define N_ATOMS 5000
#define N_PAIRS 200000
#define NF      80
#define ND      20
#define KTOT    (ND * NF)          // 1600
#define HARD_CUT 6.5f
#define PI_F 3.14159265358979323846f

typedef __attribute__((ext_vector_type(2))) float v2f;
typedef __attribute__((ext_vector_type(8))) float v8f;

__device__ __forceinline__ int lower_bound_i(const int* __restrict__ arr, int n, int key) {
    int lo = 0, hi = n;
    while (lo < hi) {
        int mid = (lo + hi) >> 1;
        if (arr[mid] < key) lo = mid + 1; else hi = mid;
    }
    return lo;
}

// ------------------------- Phase A: envsum as WMMA -------------------------

#define ENV_THREADS 160            // 5 waves; wave w owns M-tile w (rows 16w..16w+15)
#define CHUNK 64                   // pairs staged per iteration (K block)
#define CPAD  66                   // row stride: even (8B-aligned b64 frags)

__global__ __launch_bounds__(ENV_THREADS)
void envsum_wmma_kernel(const float* __restrict__ feat,
                        const float* __restrict__ dist,
                        const float* __restrict__ coord,
                        const float* __restrict__ mu,
                        const float* __restrict__ sigma,
                        const int*   __restrict__ pfirst,
                        const int*   __restrict__ psecond,
                        float* __restrict__ env,     // [N_ATOMS][KTOT]
                        float* __restrict__ envv)    // [3][N_ATOMS][KTOT]
{
    const int a    = blockIdx.x;
    const int tid  = threadIdx.x;
    const int wave = tid >> 5;
    const int lane = tid & 31;
    const int r16  = lane & 15;
    const int half = lane >> 4;
    const int kb   = half * 2;
    const int m0   = wave * 16;    // this wave's M-tile (sense-row block)

    __shared__ float s_senseT[4 * ND][CPAD];   // [m=80][k=pair-in-chunk]
    __shared__ float s_featT [NF][CPAD];       // [n=80][k=pair-in-chunk]

    const int pbeg = lower_bound_i(pfirst, N_PAIRS, a);
    const int pend = lower_bound_i(pfirst, N_PAIRS, a + 1);

    v8f acc[5] = {v8f{}, v8f{}, v8f{}, v8f{}, v8f{}};   // 5 N-tiles

    for (int base = pbeg; base < pend; base += CHUNK) {
        const int cnt = min(CHUNK, pend - base);

        // ---- stage S_all columns (one pair per thread, zero-pad tail) ----
        if (tid < CHUNK) {
            const int j = tid;
            if (j < cnt) {
                const int p = base + j;
                const float d    = dist[p];
                const float invd = 1.0f / d;
                const float cut  = (d < HARD_CUT)
                                 ? 0.5f * (__cosf(PI_F * d / HARD_CUT) + 1.0f) : 0.0f;
                const float u0 = coord[p * 3 + 0] * invd;
                const float u1 = coord[p * 3 + 1] * invd;
                const float u2 = coord[p * 3 + 2] * invd;
                #pragma unroll
                for (int s = 0; s < ND; ++s) {
                    const float t  = invd - mu[s];
                    const float sg = sigma[s];
                    const float se = __expf(-0.5f * t * t / (sg * sg)) * cut;
                    s_senseT[s][j]          = se;
                    s_senseT[ND + s][j]     = se * u0;
                    s_senseT[2 * ND + s][j] = se * u1;
                    s_senseT[3 * ND + s][j] = se * u2;
                }
            } else {
                #pragma unroll
                for (int s = 0; s < 4 * ND; ++s) s_senseT[s][j] = 0.f;
            }
        }
        // ---- stage gathered neighbor feature rows, transposed ----
        for (int idx = tid; idx < CHUNK * NF; idx += ENV_THREADS) {
            const int j = idx & (CHUNK - 1);   // pair-in-chunk
            const int f = idx >> 6;            // feature (CHUNK == 64)
            s_featT[f][j] = (j < cnt)
                          ? feat[(size_t)psecond[base + j] * NF + f] : 0.f;
        }
        __syncthreads();

        // ---- K loop over staged pairs: 5 WMMAs per step ----
        #pragma unroll
        for (int k0 = 0; k0 < CHUNK; k0 += 4) {
            const v2f aF = *(const v2f*)&s_senseT[m0 + r16][k0 + kb];
            #pragma unroll
            for (int nt = 0; nt < 5; ++nt) {
                const v2f bF = *(const v2f*)&s_featT[nt * 16 + r16][k0 + kb];
                acc[nt] = __builtin_amdgcn_wmma_f32_16x16x4_f32(
                    false, aF, false, bF, (short)0, acc[nt], false, false);
            }
        }
        __syncthreads();   // before next chunk overwrites staging
    }

    // ---- scatter accumulator tiles to workspace ----
    // C/D layout: VGPR v, lanes 0-15: M=v; lanes 16-31: M=v+8.
    #pragma unroll
    for (int nt = 0; nt < 5; ++nt) {
        const int f = nt * 16 + r16;
        #pragma unroll
        for (int v = 0; v < 8; ++v) {
            const int mrow = m0 + v + half * 8;
            const int arr  = mrow / ND;        // 0: env, 1..3: envv components
            const int s    = mrow % ND;
            float* dst = (arr == 0) ? env
                       : envv + (size_t)(arr - 1) * N_ATOMS * KTOT;
            dst[(size_t)a * KTOT + s * NF + f] = acc[nt][v];
        }
    }
}

// ------------------------- Phase B: output GEMMs ---------------------------

#define MTILES ((N_ATOMS + 15) / 16)   // 313
#define NTILES (NF / 16)               // 5

__global__ __launch_bounds__(32)
void gemm_kernel(const float* __restrict__ env,     // [N_ATOMS][KTOT]
                 const float* __restrict__ envv,    // [3][N_ATOMS][KTOT]
                 const float* __restrict__ feat,    // [N_ATOMS][NF]
                 const float* __restrict__ intw,    // [ND][NF_OUT][NF_IN]
                 const float* __restrict__ selfw,   // [NF_OUT][NF_IN]
                 const float* __restrict__ selfb,   // [NF_OUT]
                 const float* __restrict__ vecsc,   // [NF_OUT]
                 float* __restrict__ out)           // [N_ATOMS][NF]
{
    const int tile = blockIdx.x;
    const int tm = tile / NTILES;
    const int tn = tile % NTILES;
    const int m0 = tm * 16;
    const int n0 = tn * 16;

    const int lane = threadIdx.x;        // wave32
    const int r16  = lane & 15;
    const int half = lane >> 4;
    const int kb   = half * 2;           // K sub-offset for A/B fragments

    // Clamp row so padded tile rows read valid memory (never written back).
    const int arow = min(m0 + r16, N_ATOMS - 1);

    const float* __restrict__ ea  = env  + (size_t)arow * KTOT;
    const float* __restrict__ ev0 = envv + (size_t)arow * KTOT;
    const float* __restrict__ ev1 = ev0 + (size_t)N_ATOMS * KTOT;
    const float* __restrict__ ev2 = ev1 + (size_t)N_ATOMS * KTOT;

    v8f acc_s = {}, acc0 = {}, acc1 = {}, acc2 = {};

    #pragma unroll 4
    for (int k0 = 0; k0 < KTOT; k0 += 4) {
        const int k = k0 + kb;
        const int s = k / NF;
        const int f = k % NF;            // even; f,f+1 stay inside one s block

        const v2f a_s = *(const v2f*)(ea  + k);
        const v2f a0  = *(const v2f*)(ev0 + k);
        const v2f a1  = *(const v2f*)(ev1 + k);
        const v2f a2  = *(const v2f*)(ev2 + k);
        // B[k, o] = int_weights[s, o, f]; f,f+1 contiguous
        const v2f b = *(const v2f*)(intw + ((size_t)s * NF + (n0 + r16)) * NF + f);

        acc_s = __builtin_amdgcn_wmma_f32_16x16x4_f32(false, a_s, false, b, (short)0, acc_s, false, false);
        acc0  = __builtin_amdgcn_wmma_f32_16x16x4_f32(false, a0,  false, b, (short)0, acc0,  false, false);
        acc1  = __builtin_amdgcn_wmma_f32_16x16x4_f32(false, a1,  false, b, (short)0, acc1,  false, false);
        acc2  = __builtin_amdgcn_wmma_f32_16x16x4_f32(false, a2,  false, b, (short)0, acc2,  false, false);
    }

    // self-interaction: out_self[a,o] = sum_f feat[a,f] * selfw[o,f]
    const float* __restrict__ fa = feat + (size_t)arow * NF;
    #pragma unroll
    for (int k0 = 0; k0 < NF; k0 += 4) {
        const int k = k0 + kb;
        const v2f a = *(const v2f*)(fa + k);
        const v2f b = *(const v2f*)(selfw + (size_t)(n0 + r16) * NF + k);
        acc_s = __builtin_amdgcn_wmma_f32_16x16x4_f32(false, a, false, b, (short)0, acc_s, false, false);
    }

    // Epilogue: C/D layout — VGPR v, lanes 0-15: M=v; lanes 16-31: M=v+8.
    const int ocol = n0 + r16;
    const float vs   = vecsc[ocol];
    const float bias = selfb[ocol];
    #pragma unroll
    for (int v = 0; v < 8; ++v) {
        const int row = m0 + v + half * 8;
        if (row < N_ATOMS) {
            const float fv = sqrtf(acc0[v] * acc0[v] + acc1[v] * acc1[v] +
                                   acc2[v] * acc2[v] + 1e-30f);
            out[(size_t)row * NF + ocol] = acc_s[v] + vs * fv + bias;
        }
    }
}

extern "C" void kernel_launch(void* const* d_in, const int* in_sizes, int n_in,
                              void* d_out, int out_size, void* d_ws, size_t ws_size,
                              hipStream_t stream) {
    const float* feat   = (const float*)d_in[0];   // in_features  [5000*80]
    const float* dist   = (const float*)d_in[1];   // dist_pairs   [200000]
    const float* coord  = (const float*)d_in[2];   // coord_pairs  [200000*3]
    const float* intw   = (const float*)d_in[3];   // int_weights  [20*80*80]
    const float* selfw  = (const float*)d_in[4];   // selfint_w    [80*80]
    const float* selfb  = (const float*)d_in[5];   // selfint_b    [80]
    const float* vecsc  = (const float*)d_in[6];   // vecscales    [80]
    const float* mu     = (const float*)d_in[7];   // sense_mu     [20]
    const float* sigma  = (const float*)d_in[8];   // sense_sigma  [20]
    const int*   pfirst = (const int*)d_in[9];     // pair_first   [200000]
    const int*   psec   = (const int*)d_in[10];    // pair_second  [200000]
    float* out = (float*)d_out;

    float* env  = (float*)d_ws;                          // [5000][1600]
    float* envv = env + (size_t)N_ATOMS * KTOT;          // [3][5000][1600]

    envsum_wmma_kernel<<<N_ATOMS, ENV_THREADS, 0, stream>>>(
        feat, dist, coord, mu, sigma, pfirst, psec, env, envv);

    gemm_kernel<<<MTILES * NTILES, 32, 0, stream>>>(
        env, envv, feat, intw, selfw, selfb, vecsc, out);
}